// CenterLoss_37495064494859
// MI455X (gfx1250) — compile-verified
//
#include <hip/hip_runtime.h>

typedef __attribute__((ext_vector_type(2))) float v2f;
typedef __attribute__((ext_vector_type(8))) float v8f;

namespace {
constexpr int   kB        = 16384;   // batch
constexpr int   kC        = 1000;    // classes
constexpr int   kD        = 512;     // feature dim
constexpr float kClampMin = 1e-12f;
constexpr float kClampMax = 1e12f;

constexpr int kWavesPerBlock = 8;                       // 256 threads = 8 wave32
constexpr int kThreads       = kWavesPerBlock * 32;
constexpr int kRowTiles      = kB / 16;                 // 1024 tiles of 16 rows
constexpr int kBlocks        = kRowTiles / kWavesPerBlock; // 128 blocks
}

// One wave handles one 16-row tile. E = X_tile - C_gathered (16 x D).
// acc += E * E^T via V_WMMA_F32_16X16X4_F32; per-lane A and B fragments of
// (E, E^T) coincide, so a single v2f difference feeds both operands.
// Diagonal of acc = per-row squared distances.
__global__ __launch_bounds__(kThreads)
void center_loss_tile_kernel(const float* __restrict__ x,
                             const int*   __restrict__ labels,
                             const float* __restrict__ centers,
                             float*       __restrict__ partials) {
  const int lane  = threadIdx.x & 31;
  const int wave  = threadIdx.x >> 5;
  const int gwave = blockIdx.x * kWavesPerBlock + wave;   // tile index
  const int m     = lane & 15;                            // row within tile
  const int half  = lane >> 4;                            // K sub-pair select
  const int row   = gwave * 16 + m;
  const int lbl   = labels[row];

  const float* __restrict__ xr = x       + (size_t)row * kD + half * 2;
  const float* __restrict__ cr = centers + (size_t)lbl * kD + half * 2;

  v8f acc0 = {}; v8f acc1 = {}; v8f acc2 = {}; v8f acc3 = {};

  for (int k = 0; k < kD; k += 16) {
    v2f e0 = *(const v2f*)(xr + k +  0) - *(const v2f*)(cr + k +  0);
    v2f e1 = *(const v2f*)(xr + k +  4) - *(const v2f*)(cr + k +  4);
    v2f e2 = *(const v2f*)(xr + k +  8) - *(const v2f*)(cr + k +  8);
    v2f e3 = *(const v2f*)(xr + k + 12) - *(const v2f*)(cr + k + 12);
    acc0 = __builtin_amdgcn_wmma_f32_16x16x4_f32(false, e0, false, e0,
                                                 (short)0, acc0, false, false);
    acc1 = __builtin_amdgcn_wmma_f32_16x16x4_f32(false, e1, false, e1,
                                                 (short)0, acc1, false, false);
    acc2 = __builtin_amdgcn_wmma_f32_16x16x4_f32(false, e2, false, e2,
                                                 (short)0, acc2, false, false);
    acc3 = __builtin_amdgcn_wmma_f32_16x16x4_f32(false, e3, false, e3,
                                                 (short)0, acc3, false, false);
  }
  v8f acc = (acc0 + acc1) + (acc2 + acc3);

  // Diagonal extraction (16x16 f32 C/D layout):
  //   M = r        lives at lane = M        (lanes 0..7,  r = lane)
  //   M = r + 8    lives at lane = M + 16   (lanes 24..31, r = lane - 24)
  const int  r      = (lane < 16) ? lane : (lane - 24);
  const bool active = (lane < 8) || (lane >= 24);
  float dv = acc[0];
  dv = (r == 1) ? acc[1] : dv;
  dv = (r == 2) ? acc[2] : dv;
  dv = (r == 3) ? acc[3] : dv;
  dv = (r == 4) ? acc[4] : dv;
  dv = (r == 5) ? acc[5] : dv;
  dv = (r == 6) ? acc[6] : dv;
  dv = (r == 7) ? acc[7] : dv;

  float v = active ? fminf(fmaxf(dv, kClampMin), kClampMax) : 0.0f;

  // In-wave butterfly sum over the 16 clipped distances (wave32).
  #pragma unroll
  for (int off = 16; off > 0; off >>= 1)
    v += __shfl_xor(v, off, 32);

  if (lane == 0) partials[gwave] = v;
}

// Deterministic fixed-order reduction of per-wave partials.
__global__ __launch_bounds__(256)
void center_loss_reduce_kernel(const float* __restrict__ partials, int n,
                               float* __restrict__ out) {
  __shared__ float sm[256];
  float s = 0.0f;
  for (int i = threadIdx.x; i < n; i += 256) s += partials[i];
  sm[threadIdx.x] = s;
  __syncthreads();
  #pragma unroll
  for (int off = 128; off > 0; off >>= 1) {
    if ((int)threadIdx.x < off) sm[threadIdx.x] += sm[threadIdx.x + off];
    __syncthreads();
  }
  if (threadIdx.x == 0) {
    // loss = sum(clip(d_b)) / B + (C-1) * CLAMP_MIN  (masked zeros clip up)
    out[0] = sm[0] * (1.0f / (float)kB) + (float)(kC - 1) * kClampMin;
  }
}

extern "C" void kernel_launch(void* const* d_in, const int* in_sizes, int n_in,
                              void* d_out, int out_size, void* d_ws, size_t ws_size,
                              hipStream_t stream) {
  (void)in_sizes; (void)n_in; (void)out_size; (void)ws_size;
  const float* x       = (const float*)d_in[0];
  const int*   labels  = (const int*)  d_in[1];
  const float* centers = (const float*)d_in[2];
  float*       out     = (float*)d_out;
  float*       partials = (float*)d_ws;   // kRowTiles floats = 4 KB scratch

  center_loss_tile_kernel<<<kBlocks, kThreads, 0, stream>>>(x, labels, centers,
                                                            partials);
  center_loss_reduce_kernel<<<1, 256, 0, stream>>>(partials, kRowTiles, out);
}